// RGAT_39505109188795
// MI455X (gfx1250) — compile-verified
//
#include <hip/hip_runtime.h>
#include <hip/hip_bf16.h>

// ---------------------------------------------------------------------------
// RGAT (2 layers, R=8 relations, F=128) for MI455X / gfx1250 (wave32, WMMA).
//
// Per layer:
//   1) xw[r] = H @ W[r]      -> FP32 WMMA 16x16x4 (LDS-staged, pair-swizzled W[r])
//   2) logits[e] = leakyrelu(xw[et,dst].q + xw[et,src].k)  (wave-per-edge dot)
//      segment max via float atomicMax
//   3) e = exp(logit - m[dst]); denom[dst] += e
//   4) agg[dst] += (e/denom[dst]) * xw[et,src]  (wave-per-edge scatter-add)
//   5) out = relu(agg + bias)
// ---------------------------------------------------------------------------

#define FDIM 128

typedef float v2f __attribute__((ext_vector_type(2)));
typedef float v8f __attribute__((ext_vector_type(8)));

// ---------------------------------------------------------------------------
// Zero / init buffers used by one layer pass.
__global__ void rgat_init(float* __restrict__ agg, float* __restrict__ m,
                          float* __restrict__ denom, int N, int total) {
    int i = blockIdx.x * blockDim.x + threadIdx.x;
    if (i < total) agg[i] = 0.0f;
    if (i < N) {
        m[i] = -__builtin_inff();
        denom[i] = 0.0f;
    }
}

// ---------------------------------------------------------------------------
// xw[r] = H @ W[r].  grid = (ceil(N/128), R), block = 256 (8 waves).
// Each wave computes a 16x128 output slice with FP32 WMMA 16x16x4.
//
// LDS holds W[r] "pair-swizzled": Bs2[k/2][col] = {W[k][col], W[k+1][col]}
// so a B fragment (K-pair x 16 columns) is one aligned ds_load_b64 per lane,
// landing directly in an even-aligned VGPR pair for WMMA SRC1.
__global__ __launch_bounds__(256) void rgat_gemm(const float* __restrict__ H,
                                                 const float* __restrict__ Wall,
                                                 float* __restrict__ XW, int N) {
    __shared__ v2f Bs2[(FDIM / 2) * FDIM];  // 64 KB

    const int r = blockIdx.y;
    const float* Wr = Wall + (size_t)r * FDIM * FDIM;
    for (int i = threadIdx.x; i < FDIM * FDIM / 4; i += 256) {
        const int k = i >> 5;            // source row (128/4 float4 per row)
        const int c0 = (i & 31) * 4;     // source column of this float4
        const float4 w4 = ((const float4*)Wr)[i];
        float* d = ((float*)&Bs2[(k >> 1) * FDIM + c0]) + (k & 1);
        d[0] = w4.x; d[2] = w4.y; d[4] = w4.z; d[6] = w4.w;
    }
    __syncthreads();

    const int wave = threadIdx.x >> 5;
    const int lane = threadIdx.x & 31;
    const int m0 = blockIdx.x * 128 + wave * 16;   // wave-uniform
    if (m0 >= N) return;                           // whole-wave exit: EXEC stays full

    const int half = lane >> 4;    // 0: K pair {k,k+1}; 1: K pair {k+2,k+3}
    const int l16 = lane & 15;
    const float* Arow = H + (size_t)(m0 + l16) * FDIM;

    v8f acc[8] = {};
    for (int k = 0; k < FDIM; k += 4) {
        // A fragment: 16x4 f32, lane l16 = row, (a.x,a.y) = K=ka,ka+1
        v2f a = *(const v2f*)(Arow + k + half * 2);
        const v2f* Brow = Bs2 + ((k >> 1) + half) * FDIM;
#pragma unroll
        for (int n = 0; n < 8; ++n) {
            // B fragment: 4x16 f32, lane l16 = column, one b64 load
            v2f b = Brow[n * 16 + l16];
            acc[n] = __builtin_amdgcn_wmma_f32_16x16x4_f32(
                false, a, false, b, (short)0, acc[n], false, false);
        }
    }

    // C/D layout: VGPR v, lanes 0-15 -> M = v, lanes 16-31 -> M = v+8; N = l16.
    float* outBase = XW + ((size_t)r * N + m0) * FDIM;
#pragma unroll
    for (int n = 0; n < 8; ++n) {
#pragma unroll
        for (int v = 0; v < 8; ++v) {
            outBase[(size_t)(v + half * 8) * FDIM + n * 16 + l16] = acc[n][v];
        }
    }
}

// ---------------------------------------------------------------------------
// Wave per edge: logits + segment max.
__global__ __launch_bounds__(256) void rgat_edge_logits(
    const float* __restrict__ xw, const int* __restrict__ src,
    const int* __restrict__ dst, const int* __restrict__ et,
    const float* __restrict__ q, const float* __restrict__ k,
    float* __restrict__ logits, float* __restrict__ m, int N, int E) {
    const int wave = threadIdx.x >> 5;
    const int lane = threadIdx.x & 31;
    const int e = blockIdx.x * 8 + wave;       // wave-uniform
    if (e >= E) return;

    const int r = et[e], s = src[e], d = dst[e];
    const float4 oi = ((const float4*)(xw + ((size_t)r * N + d) * FDIM))[lane];
    const float4 oj = ((const float4*)(xw + ((size_t)r * N + s) * FDIM))[lane];
    const float4 qv = ((const float4*)q)[lane];
    const float4 kv = ((const float4*)k)[lane];

    float p = oi.x * qv.x + oi.y * qv.y + oi.z * qv.z + oi.w * qv.w
            + oj.x * kv.x + oj.y * kv.y + oj.z * kv.z + oj.w * kv.w;
#pragma unroll
    for (int off = 16; off > 0; off >>= 1)
        p += __shfl_xor(p, off, 32);

    if (lane == 0) {
        float lg = p > 0.0f ? p : 0.2f * p;    // leaky_relu, slope 0.2
        logits[e] = lg;
        atomicMax(&m[d], lg);
    }
}

// ---------------------------------------------------------------------------
// Thread per edge: exp(logit - m[dst]) and denominator accumulation.
__global__ void rgat_edge_exp(const float* __restrict__ logits,
                              const float* __restrict__ m,
                              const int* __restrict__ dst,
                              float* __restrict__ ee, float* __restrict__ denom,
                              int E) {
    int e = blockIdx.x * blockDim.x + threadIdx.x;
    if (e >= E) return;
    int d = dst[e];
    float v = __expf(logits[e] - m[d]);
    ee[e] = v;
    atomicAdd(&denom[d], v);
}

// ---------------------------------------------------------------------------
// Wave per edge: agg[dst] += alpha * xw[et, src].
__global__ __launch_bounds__(256) void rgat_edge_agg(
    const float* __restrict__ xw, const int* __restrict__ src,
    const int* __restrict__ dst, const int* __restrict__ et,
    const float* __restrict__ ee, const float* __restrict__ denom,
    float* __restrict__ agg, int N, int E) {
    const int wave = threadIdx.x >> 5;
    const int lane = threadIdx.x & 31;
    const int e = blockIdx.x * 8 + wave;
    if (e >= E) return;

    const int r = et[e], s = src[e], d = dst[e];
    const float alpha = ee[e] / (denom[d] + 1e-16f);
    const float4 oj = ((const float4*)(xw + ((size_t)r * N + s) * FDIM))[lane];
    float* ap = agg + (size_t)d * FDIM + lane * 4;
    atomicAdd(ap + 0, alpha * oj.x);
    atomicAdd(ap + 1, alpha * oj.y);
    atomicAdd(ap + 2, alpha * oj.z);
    atomicAdd(ap + 3, alpha * oj.w);
}

// ---------------------------------------------------------------------------
__global__ void rgat_bias_relu(const float* __restrict__ agg,
                               const float* __restrict__ b,
                               float* __restrict__ out, int total) {
    int i = blockIdx.x * blockDim.x + threadIdx.x;
    if (i >= total) return;
    float v = agg[i] + b[i & (FDIM - 1)];
    out[i] = v > 0.0f ? v : 0.0f;
}

// ---------------------------------------------------------------------------
extern "C" void kernel_launch(void* const* d_in, const int* in_sizes, int n_in,
                              void* d_out, int out_size, void* d_ws, size_t ws_size,
                              hipStream_t stream) {
    const float* x      = (const float*)d_in[0];
    const float* W      = (const float*)d_in[1];
    const float* att_q  = (const float*)d_in[2];
    const float* att_k  = (const float*)d_in[3];
    const float* bias   = (const float*)d_in[4];
    const int* edge_idx = (const int*)d_in[5];
    const int* edge_typ = (const int*)d_in[6];

    const int F = FDIM;
    const int N = in_sizes[0] / F;                 // 50000
    const int E = in_sizes[6];                     // 500000
    const int L = in_sizes[4] / F;                 // 2
    const int R = in_sizes[1] / (L * F * F);       // 8

    const int* src = edge_idx;
    const int* dst = edge_idx + E;

    // Workspace carve-out (floats).
    float* ws = (float*)d_ws;
    float* xw     = ws; ws += (size_t)R * N * F;   // 204.8 MB
    float* agg    = ws; ws += (size_t)N * F;
    float* hbuf   = ws; ws += (size_t)N * F;
    float* logits = ws; ws += (size_t)E;
    float* ee     = ws; ws += (size_t)E;
    float* m      = ws; ws += (size_t)N;
    float* denom  = ws; ws += (size_t)N;

    for (int l = 0; l < L; ++l) {
        const float* h  = (l == 0) ? x : hbuf;
        float* out      = (l == L - 1) ? (float*)d_out : hbuf;
        const float* Wl = W     + (size_t)l * R * F * F;
        const float* ql = att_q + (size_t)l * F;
        const float* kl = att_k + (size_t)l * F;
        const float* bl = bias  + (size_t)l * F;

        const int total = N * F;
        rgat_init<<<(total + 255) / 256, 256, 0, stream>>>(agg, m, denom, N, total);

        dim3 ggrid((N + 127) / 128, R);
        rgat_gemm<<<ggrid, 256, 0, stream>>>(h, Wl, xw, N);

        rgat_edge_logits<<<(E + 7) / 8, 256, 0, stream>>>(xw, src, dst, edge_typ,
                                                          ql, kl, logits, m, N, E);
        rgat_edge_exp<<<(E + 255) / 256, 256, 0, stream>>>(logits, m, dst, ee, denom, E);
        rgat_edge_agg<<<(E + 7) / 8, 256, 0, stream>>>(xw, src, dst, edge_typ,
                                                       ee, denom, agg, N, E);
        rgat_bias_relu<<<(total + 255) / 256, 256, 0, stream>>>(agg, bl, out, total);
    }
}